// SpectralLinearFilter_8014408974863
// MI455X (gfx1250) — compile-verified
//
#include <hip/hip_runtime.h>
#include <hip/hip_bf16.h>

// ---------------------------------------------------------------------------
// SpectralLinearFilter for MI455X (gfx1250, wave32).
// Pipeline: hist -> scan -> padzero -> scatter(compaction) -> WMMA syrk
//           -> reduce + Cholesky solve -> per-pixel score.
// Fully deterministic: no atomics; counting-sort compaction with stable ranks;
// fixed-order partial reductions.
// ---------------------------------------------------------------------------

typedef float v2f __attribute__((ext_vector_type(2)));
typedef float v8f __attribute__((ext_vector_type(8)));

#define NB 2
#define NC 64
#define NPIX 147456            // 384*384
#define NCLS 20
#define BLKPX 1024             // pixels per histogram/scatter block (one wave)
#define CHUNKS (BLKPX / 32)
#define NBLKS (NPIX / BLKPX)   // 144
#define PARTS 16               // waves per (b,k) in the syrk
#define PADALIGN (PARTS * 4)   // class region padded so each part is mult of 4
#define NPAD_MAX (NPIX + NCLS * PADALIGN)
#define MIN_CLASS_SIZE 5000
#define EPSV 1e-6f

// ---------------------------------------------------------------------------
// Kernel 1: per-block per-class histogram (wave32 ballots, no atomics)
// ---------------------------------------------------------------------------
__global__ __launch_bounds__(32) void hist_kernel(const int* __restrict__ seg,
                                                  int* __restrict__ hist) {
  int blk = blockIdx.x;              // b * NBLKS + bb
  int b = blk / NBLKS, bb = blk % NBLKS;
  int lane = threadIdx.x;
  const int* s = seg + (size_t)b * NPIX + (size_t)bb * BLKPX;
  int cnt = 0;                       // lane l accumulates count of class l
  for (int ch = 0; ch < CHUNKS; ++ch) {
    int myk = s[ch * 32 + lane];
#pragma unroll
    for (int k = 0; k < NCLS; ++k) {
      unsigned bal = __builtin_amdgcn_ballot_w32(myk == k);
      if (lane == k) cnt += __popc(bal);
    }
  }
  if (lane < NCLS) hist[(size_t)blk * NCLS + lane] = cnt;
}

// ---------------------------------------------------------------------------
// Kernel 2: serial scans (tiny): per-(b,k) block offsets, class bases, pads
// classinfo[(b*NCLS+k)*3 + {0,1,2}] = {base_row (batch-rel), count, padlen}
// ---------------------------------------------------------------------------
__global__ __launch_bounds__(64) void scan_kernel(const int* __restrict__ hist,
                                                  int* __restrict__ blockoff,
                                                  int* __restrict__ classinfo) {
  int tid = threadIdx.x;
  __shared__ int tot[NB * NCLS];
  __shared__ int padl[NB * NCLS];
  if (tid < NB * NCLS) {
    int b = tid / NCLS, k = tid % NCLS;
    int run = 0;
    for (int blk = 0; blk < NBLKS; ++blk) {
      size_t idx = ((size_t)(b * NBLKS + blk)) * NCLS + k;
      blockoff[idx] = run;
      run += hist[idx];
    }
    tot[tid] = run;
    padl[tid] = (run + PADALIGN - 1) & ~(PADALIGN - 1);
  }
  __syncthreads();
  if (tid < NB) {
    int b = tid, base = 0;
    for (int k = 0; k < NCLS; ++k) {
      classinfo[(b * NCLS + k) * 3 + 0] = base;
      classinfo[(b * NCLS + k) * 3 + 1] = tot[b * NCLS + k];
      classinfo[(b * NCLS + k) * 3 + 2] = padl[b * NCLS + k];
      base += padl[b * NCLS + k];
    }
  }
}

// ---------------------------------------------------------------------------
// Kernel 3: zero the padding rows of the compacted buffer (ws is poisoned).
// Zero rows contribute nothing to X^T X or to column sums.
// ---------------------------------------------------------------------------
__global__ __launch_bounds__(256) void padzero_kernel(const int* __restrict__ classinfo,
                                                      float* __restrict__ comp) {
  int bk = blockIdx.x;
  int b = bk / NCLS;
  int base = classinfo[bk * 3 + 0];
  int count = classinfo[bk * 3 + 1];
  int padl = classinfo[bk * 3 + 2];
  int nel = (padl - count) * NC;
  float* dst = comp + ((size_t)b * NPAD_MAX + base + count) * NC;
  for (int i = threadIdx.x; i < nel; i += 256) dst[i] = 0.0f;
}

// ---------------------------------------------------------------------------
// Kernel 4: stable counting-sort scatter. One wave per 1024-pixel block.
// Reads channel-planes coalesced, transposes through LDS, writes compacted
// pixel rows (64 contiguous floats) coalesced.
// ---------------------------------------------------------------------------
__global__ __launch_bounds__(32) void scatter_kernel(const float* __restrict__ hyper,
                                                     const int* __restrict__ seg,
                                                     const int* __restrict__ blockoff,
                                                     const int* __restrict__ classinfo,
                                                     float* __restrict__ comp) {
  int blk = blockIdx.x;
  int b = blk / NBLKS, bb = blk % NBLKS;
  int lane = threadIdx.x;
  __shared__ float stage[32 * 65];   // stage[p][c], stride 65 -> conflict-free
  __shared__ int posArr[32];
  int run = 0;                       // lane l = running count of class l
  unsigned ltmask = (1u << lane) - 1u;
  size_t nbase0 = (size_t)bb * BLKPX;

  for (int ch = 0; ch < CHUNKS; ++ch) {
    size_t nbase = nbase0 + (size_t)ch * 32;
    int n = (int)(nbase + lane);
    int myk = seg[(size_t)b * NPIX + n];
    unsigned balmy = 0;
    int pend = 0;
#pragma unroll
    for (int k = 0; k < NCLS; ++k) {
      unsigned bal = __builtin_amdgcn_ballot_w32(myk == k);
      if (myk == k) balmy = bal;
      if (lane == k) pend = __popc(bal);
    }
    int runbase = __shfl(run, myk, 32);      // class-l running count, pre-update
    int myrank = __popc(balmy & ltmask);     // stable rank within this 32-chunk
    run += (lane < NCLS) ? pend : 0;

    int cbase = classinfo[(b * NCLS + myk) * 3 + 0];
    int brel = blockoff[((size_t)(b * NBLKS + bb)) * NCLS + myk];
    int pos = cbase + brel + runbase + myrank;
    posArr[lane] = pos;

    // gather this lane's pixel vector, transposed via LDS (coalesced reads)
    for (int c = 0; c < NC; ++c)
      stage[lane * 65 + c] = hyper[((size_t)b * NC + c) * NPIX + n];
    __syncthreads();
    // write 32 compacted rows, each row coalesced (2x128B per row)
    for (int p = 0; p < 32; ++p) {
      size_t dst = ((size_t)b * NPAD_MAX + posArr[p]) * NC;
      comp[dst + lane] = stage[p * 65 + lane];
      comp[dst + lane + 32] = stage[p * 65 + lane + 32];
    }
    __syncthreads();
  }
}

// ---------------------------------------------------------------------------
// Kernel 5: WMMA fp32 syrk. One wave per (part,b,k). Computes a partial
// Sxx = X_k^T X_k (64x64) as 16 tiles of v_wmma_f32_16x16x4_f32, plus partial
// column sums. Padding rows are zero -> contribute nothing.
//
// fp32 fragment layouts (ISA 7.12.2):
//   A 16x4: lane l holds rows M=l&15, K = 2*(l>>4) + {reg0,reg1}
//   B 4x16: reg j, lane l holds row K = j + 2*(l>>4), col N = l&15
// With comp row-major (pixel rows x 64 ch), both fragments for channel-block
// `blk` reduce to the SAME per-lane loads -> one fragment array F[4] feeds
// all 16 accumulators.
// ---------------------------------------------------------------------------
__global__ __launch_bounds__(32) void syrk_kernel(const float* __restrict__ comp,
                                                  const int* __restrict__ classinfo,
                                                  float* __restrict__ sxxpart,
                                                  float* __restrict__ sumpart) {
  int gid = blockIdx.x;                    // (part*NB + b)*NCLS + k
  int part = gid / (NB * NCLS);
  int bk = gid % (NB * NCLS);
  int b = bk / NCLS;
  int lane = threadIdx.x;

  int base = classinfo[bk * 3 + 0];
  int padl = classinfo[bk * 3 + 2];
  int plen = padl / PARTS;                 // multiple of 4 by construction
  size_t rowbase = (size_t)b * NPAD_MAX + base + (size_t)part * plen;

  int m = lane & 15;
  int kb = (lane >> 4) << 1;               // 0 or 2

  v8f acc[4][4] = {};
  for (int p = 0; p < plen; p += 4) {
    const float* row0 = comp + (rowbase + p + kb) * NC;
    const float* row1 = row0 + NC;
    v2f F[4];
#pragma unroll
    for (int blk = 0; blk < 4; ++blk) {
      F[blk].x = row0[blk * 16 + m];
      F[blk].y = row1[blk * 16 + m];
    }
#pragma unroll
    for (int cb = 0; cb < 4; ++cb)
#pragma unroll
      for (int db = 0; db < 4; ++db)
        acc[cb][db] = __builtin_amdgcn_wmma_f32_16x16x4_f32(
            false, F[cb], false, F[db], (short)0, acc[cb][db], false, false);
  }

  // store partial Sxx (C/D layout: reg j -> row j + 8*(lane>=16), col = lane&15)
  size_t ob = (size_t)gid * (NC * NC);
  int col = lane & 15;
  int rof = (lane >> 4) * 8;
#pragma unroll
  for (int cb = 0; cb < 4; ++cb)
#pragma unroll
    for (int db = 0; db < 4; ++db)
#pragma unroll
      for (int j = 0; j < 8; ++j) {
        int r = cb * 16 + rof + j;
        sxxpart[ob + (size_t)r * NC + db * 16 + col] = acc[cb][db][j];
      }

  // partial column sums (second streaming pass; hits L2)
  float s0 = 0.0f, s1 = 0.0f;
  for (int p = 0; p < plen; ++p) {
    const float* row = comp + (rowbase + p) * NC;
    s0 += row[lane];
    s1 += row[lane + 32];
  }
  sumpart[(size_t)gid * NC + lane] = s0;
  sumpart[(size_t)gid * NC + lane + 32] = s1;
}

// ---------------------------------------------------------------------------
// Kernel 6: fixed-order partial reduction, cov build, in-LDS Cholesky,
// triangular solves for w = cov^-1 p, and per-class scalars.
// params[(b*NCLS+k)*66 + {0..63,64,65}] = {w[64], offset = mu.w, scale}
// ---------------------------------------------------------------------------
__global__ __launch_bounds__(256) void reduce_solve_kernel(const float* __restrict__ sxxpart,
                                                           const float* __restrict__ sumpart,
                                                           const int* __restrict__ classinfo,
                                                           const float* __restrict__ pattern,
                                                           float* __restrict__ params) {
  int bk = blockIdx.x;
  int tid = threadIdx.x;
  __shared__ float covL[64 * 65];          // stride 65: bank-conflict-free
  __shared__ float meanL[64], pL[64], yv[64], wv[64];

  int count = classinfo[bk * 3 + 1];
  float safe = (float)(count > 0 ? count : 1);

  if (tid < 64) {
    float s = 0.0f;
    for (int part = 0; part < PARTS; ++part)
      s += sumpart[((size_t)part * NB * NCLS + bk) * NC + tid];
    meanL[tid] = s / safe;
    pL[tid] = pattern[tid];
  }
  __syncthreads();

  for (int idx = tid; idx < NC * NC; idx += 256) {
    int i = idx >> 6, j = idx & 63;
    float s = 0.0f;
    for (int part = 0; part < PARTS; ++part)
      s += sxxpart[((size_t)part * NB * NCLS + bk) * (NC * NC) + idx];
    float cv = s / safe - meanL[i] * meanL[j];
    if (i == j) cv += EPSV;
    covL[i * 65 + j] = cv;
  }
  __syncthreads();

  // Cholesky, lower triangle in covL
  for (int kk = 0; kk < NC; ++kk) {
    if (tid == 0) covL[kk * 65 + kk] = sqrtf(covL[kk * 65 + kk]);
    __syncthreads();
    float d = covL[kk * 65 + kk];
    for (int i = kk + 1 + tid; i < NC; i += 256) covL[i * 65 + kk] /= d;
    __syncthreads();
    int nrem = NC - 1 - kk;
    for (int t = tid; t < nrem * nrem; t += 256) {
      int i = kk + 1 + t / nrem;
      int j = kk + 1 + t % nrem;
      if (j <= i) covL[i * 65 + j] -= covL[i * 65 + kk] * covL[j * 65 + kk];
    }
    __syncthreads();
  }

  // forward solve L y = p
  if (tid < 64) yv[tid] = pL[tid];
  __syncthreads();
  for (int j = 0; j < NC; ++j) {
    if (tid == 0) yv[j] /= covL[j * 65 + j];
    __syncthreads();
    float yj = yv[j];
    for (int i = j + 1 + tid; i < NC; i += 256) yv[i] -= covL[i * 65 + j] * yj;
    __syncthreads();
  }
  // backward solve L^T w = y
  if (tid < 64) wv[tid] = yv[tid];
  __syncthreads();
  for (int j = NC - 1; j >= 0; --j) {
    if (tid == 0) wv[j] /= covL[j * 65 + j];
    __syncthreads();
    float wj = wv[j];
    for (int i = tid; i < j; i += 256) wv[i] -= covL[j * 65 + i] * wj;
    __syncthreads();
  }

  if (tid < 64) params[(size_t)bk * 66 + tid] = wv[tid];
  if (tid == 0) {
    float pw = 0.0f, off = 0.0f;
    for (int c = 0; c < NC; ++c) {
      pw += pL[c] * wv[c];
      off += meanL[c] * wv[c];
    }
    float denom = sqrtf(pw);
    float scale = (count >= MIN_CLASS_SIZE && denom > 0.0f) ? (1.0f / denom) : 0.0f;
    params[(size_t)bk * 66 + 64] = off;
    params[(size_t)bk * 66 + 65] = scale;
  }
}

// ---------------------------------------------------------------------------
// Kernel 7: per-pixel score = (x.w_k - offset_k) * scale_k. Coalesced channel
// loads, per-class w table in LDS (stride 65 -> conflict-free gathers).
// ---------------------------------------------------------------------------
__global__ __launch_bounds__(256) void score_kernel(const float* __restrict__ hyper,
                                                    const int* __restrict__ seg,
                                                    const float* __restrict__ params,
                                                    float* __restrict__ out) {
  int b = blockIdx.y;
  int tid = threadIdx.x;
  __shared__ float wS[NCLS * 65];
  __shared__ float offS[NCLS], sclS[NCLS];
  for (int i = tid; i < NCLS * NC; i += 256) {
    int k = i >> 6, c = i & 63;
    wS[k * 65 + c] = params[((size_t)b * NCLS + k) * 66 + c];
  }
  if (tid < NCLS) {
    offS[tid] = params[((size_t)b * NCLS + tid) * 66 + 64];
    sclS[tid] = params[((size_t)b * NCLS + tid) * 66 + 65];
  }
  __syncthreads();

  size_t n = (size_t)blockIdx.x * 256 + tid;     // grid covers NPIX exactly
  int k = seg[(size_t)b * NPIX + n];
  const float* xp = hyper + (size_t)b * NC * NPIX + n;
  float acc = 0.0f;
#pragma unroll
  for (int c = 0; c < NC; ++c)
    acc = fmaf(xp[(size_t)c * NPIX], wS[k * 65 + c], acc);
  out[(size_t)b * NPIX + n] = (acc - offS[k]) * sclS[k];
}

// ---------------------------------------------------------------------------
extern "C" void kernel_launch(void* const* d_in, const int* in_sizes, int n_in,
                              void* d_out, int out_size, void* d_ws, size_t ws_size,
                              hipStream_t stream) {
  const float* hyper = (const float*)d_in[0];
  const float* pattern = (const float*)d_in[1];
  const int* seg = (const int*)d_in[2];
  float* out = (float*)d_out;

  char* ws = (char*)d_ws;
  auto al = [](size_t x) { return (x + 255) & ~(size_t)255; };
  size_t off = 0;
  float* comp = (float*)(ws + off);
  off += al((size_t)NB * NPAD_MAX * NC * sizeof(float));
  int* hist = (int*)(ws + off);
  off += al((size_t)NB * NBLKS * NCLS * sizeof(int));
  int* blockoff = (int*)(ws + off);
  off += al((size_t)NB * NBLKS * NCLS * sizeof(int));
  int* classinfo = (int*)(ws + off);
  off += al((size_t)NB * NCLS * 3 * sizeof(int));
  float* sxxpart = (float*)(ws + off);
  off += al((size_t)PARTS * NB * NCLS * NC * NC * sizeof(float));
  float* sumpart = (float*)(ws + off);
  off += al((size_t)PARTS * NB * NCLS * NC * sizeof(float));
  float* params = (float*)(ws + off);
  off += al((size_t)NB * NCLS * 66 * sizeof(float));

  hipLaunchKernelGGL(hist_kernel, dim3(NB * NBLKS), dim3(32), 0, stream, seg, hist);
  hipLaunchKernelGGL(scan_kernel, dim3(1), dim3(64), 0, stream, hist, blockoff, classinfo);
  hipLaunchKernelGGL(padzero_kernel, dim3(NB * NCLS), dim3(256), 0, stream, classinfo, comp);
  hipLaunchKernelGGL(scatter_kernel, dim3(NB * NBLKS), dim3(32), 0, stream, hyper, seg,
                     blockoff, classinfo, comp);
  hipLaunchKernelGGL(syrk_kernel, dim3(PARTS * NB * NCLS), dim3(32), 0, stream, comp,
                     classinfo, sxxpart, sumpart);
  hipLaunchKernelGGL(reduce_solve_kernel, dim3(NB * NCLS), dim3(256), 0, stream, sxxpart,
                     sumpart, classinfo, pattern, params);
  hipLaunchKernelGGL(score_kernel, dim3(NPIX / 256, NB), dim3(256), 0, stream, hyper, seg,
                     params, out);
}